// RepetitionAwareLoss_28759101014359
// MI455X (gfx1250) — compile-verified
//
#include <hip/hip_runtime.h>

#define B_DIM 8
#define S_DIM 512
#define V_DIM 32000
#define ROWS  (B_DIM * S_DIM)
#define V4    (V_DIM / 4)          // 8000 float4 per row
#define EPS_LS 0.1f
#define REP_W  0.2f
#define IGN    (-100)

// ---- CDNA5 async global->LDS path (guarded; falls back to direct loads) ----
#if defined(__AMDGCN__) && __has_builtin(__builtin_amdgcn_global_load_async_to_lds_b128)
#define HAVE_ASYNC 1
#else
#define HAVE_ASYNC 0
#endif

#if HAVE_ASYNC
typedef int v4i __attribute__((vector_size(16)));
typedef __attribute__((address_space(1))) v4i* as1_v4i_p;   // global b128 pointer
typedef __attribute__((address_space(3))) v4i* as3_v4i_p;   // LDS b128 pointer
#if __has_builtin(__builtin_amdgcn_s_wait_asynccnt)
#define WAIT_ASYNC(n) __builtin_amdgcn_s_wait_asynccnt(n)
#else
#define WAIT_ASYNC(n) asm volatile("s_wait_asynccnt %0" ::"i"(n) : "memory")
#endif
#endif

// ------------------------------- kernels ------------------------------------
// row_kernel first so the disasm snippet shows its async-load prologue.

__global__ __launch_bounds__(256) void row_kernel(const float* __restrict__ logits,
                                                  const int* __restrict__ labels,
                                                  float* __restrict__ rowLoss,
                                                  float* __restrict__ rowValid,
                                                  unsigned* __restrict__ counts) {
    const int row = blockIdx.x;
    const int tid = threadIdx.x;
    const float* gbase = logits + (size_t)row * V_DIM;
    const float4* g4 = (const float4*)gbase;

    // per-thread streaming state: online logsumexp, plain sum, argmax
    float m = -__builtin_inff();   // running max for logsumexp
    float s = 0.0f;                // sum of exp(x - m)
    float sum = 0.0f;              // sum of x
    float av = -__builtin_inff();  // argmax value
    int   ai = 0x7fffffff;         // argmax index

    auto acc = [&](float x, int idx) {
        sum += x;
        if (x > av) { av = x; ai = idx; }
        if (x <= m) {
            s += __expf(x - m);
        } else {
            s = s * __expf(m - x) + 1.0f;
            m = x;
        }
    };

#if HAVE_ASYNC
    constexpr int CH4 = 1024;                       // float4 per chunk (16 KB)
    constexpr int NCH = (V4 + CH4 - 1) / CH4;       // 8 chunks
    __shared__ float4 buf[2][CH4];

    auto stage = [&](int c) {
        const int cb = c & 1;
#pragma unroll
        for (int k = 0; k < 4; ++k) {               // exactly 4 issues per wave
            const int slot = k * 256 + tid;
            int gf4 = c * CH4 + slot;
            if (gf4 > V4 - 1) gf4 = V4 - 1;         // clamp: keep issue count uniform
            __builtin_amdgcn_global_load_async_to_lds_b128(
                (as1_v4i_p)(g4 + gf4), (as3_v4i_p)(&buf[cb][slot]), 0, 0);
        }
    };
    auto compute = [&](int c) {
        const int cb = c & 1;
        int lim = V4 - c * CH4; if (lim > CH4) lim = CH4;
        for (int j = tid; j < lim; j += 256) {
            float4 x = buf[cb][j];
            const int base = (c * CH4 + j) * 4;
            acc(x.x, base); acc(x.y, base + 1); acc(x.z, base + 2); acc(x.w, base + 3);
        }
    };

    stage(0);
    for (int c = 0; c < NCH - 1; ++c) {
        stage(c + 1);          // prefetch next chunk into the other buffer
        WAIT_ASYNC(4);         // wait until only next chunk's 4 issues outstanding
        __syncthreads();       // all waves' portions of chunk c landed
        compute(c);
        __syncthreads();       // chunk c buffer free for reuse
    }
    WAIT_ASYNC(0);
    __syncthreads();
    compute(NCH - 1);
#else
    for (int j = tid; j < V4; j += 256) {
        if (j + 1024 < V4) __builtin_prefetch(g4 + j + 1024, 0, 0);  // global_prefetch_b8
        float4 x = g4[j];
        const int base = j * 4;
        acc(x.x, base); acc(x.y, base + 1); acc(x.z, base + 2); acc(x.w, base + 3);
    }
#endif

    // ---- wave32 butterfly reduce ----
    for (int mask = 16; mask; mask >>= 1) {
        float om = __shfl_xor(m, mask, 32);
        float os = __shfl_xor(s, mask, 32);
        float nm = fmaxf(m, om);
        s = s * __expf(m - nm) + os * __expf(om - nm);
        m = nm;
        sum += __shfl_xor(sum, mask, 32);
        float oav = __shfl_xor(av, mask, 32);
        int   oai = __shfl_xor(ai, mask, 32);
        if (oav > av || (oav == av && oai < ai)) { av = oav; ai = oai; }
    }

    // ---- cross-wave merge (8 waves) ----
    __shared__ float rm[8], rs[8], rsum[8], rav[8];
    __shared__ int   rai[8];
    if ((tid & 31) == 0) {
        const int w = tid >> 5;
        rm[w] = m; rs[w] = s; rsum[w] = sum; rav[w] = av; rai[w] = ai;
    }
    __syncthreads();
    if (tid == 0) {
        m = rm[0]; s = rs[0]; sum = rsum[0]; av = rav[0]; ai = rai[0];
        for (int w = 1; w < 8; ++w) {
            float nm = fmaxf(m, rm[w]);
            s = s * __expf(m - nm) + rs[w] * __expf(rm[w] - nm);
            m = nm;
            sum += rsum[w];
            if (rav[w] > av || (rav[w] == av && rai[w] < ai)) { av = rav[w]; ai = rai[w]; }
        }
        const int lbl = labels[row];
        const int valid = (lbl != IGN);
        const float lg = gbase[valid ? lbl : 0];
        const float logZ = m + logf(s);
        const float nll = logZ - lg;                       // -log p(label)
        const float smooth = logZ - sum * (1.0f / V_DIM);  // -mean(log p)
        const float tok = (1.0f - EPS_LS) * nll + EPS_LS * smooth;
        rowLoss[row]  = valid ? tok : 0.0f;
        rowValid[row] = valid ? 1.0f : 0.0f;
        if (valid) atomicAdd(&counts[(row / S_DIM) * V_DIM + ai], 1u);
    }
}

__global__ __launch_bounds__(256) void zero_counts_kernel(unsigned* counts) {
    int i = blockIdx.x * 256 + threadIdx.x;
    if (i < B_DIM * V_DIM) counts[i] = 0u;
}

__global__ __launch_bounds__(256) void rep_kernel(const unsigned* __restrict__ counts,
                                                  float* __restrict__ perB) {
    const int b = blockIdx.x, tid = threadIdx.x;
    const unsigned* c = counts + (size_t)b * V_DIM;
    __shared__ float sf[256];

    // pass 1: total
    float t = 0.0f;
    for (int v = tid; v < V_DIM; v += 256) t += (float)c[v];
    sf[tid] = t; __syncthreads();
    for (int o = 128; o; o >>= 1) { if (tid < o) sf[tid] += sf[tid + o]; __syncthreads(); }
    const float T = sf[0];
    __syncthreads();

    // pass 2: entropy terms + unique count (matches reference's log(p + 1e-10))
    const float inv = 1.0f / fmaxf(T, 1.0f);
    float ent = 0.0f, uniq = 0.0f;
    for (int v = tid; v < V_DIM; v += 256) {
        const unsigned cv = c[v];
        if (cv) {
            const float p = (float)cv * inv;
            ent += p * logf(p + 1e-10f);
            uniq += 1.0f;
        }
    }
    sf[tid] = ent; __syncthreads();
    for (int o = 128; o; o >>= 1) { if (tid < o) sf[tid] += sf[tid + o]; __syncthreads(); }
    ent = sf[0]; __syncthreads();
    sf[tid] = uniq; __syncthreads();
    for (int o = 128; o; o >>= 1) { if (tid < o) sf[tid] += sf[tid + o]; __syncthreads(); }
    if (tid == 0) {
        const float entropy = -ent;
        const float maxent = logf(sf[0] + 1.0f);
        perB[b] = (T > 0.0f) ? (1.0f - entropy / maxent) : 0.0f;
    }
}

__global__ __launch_bounds__(256) void final_kernel(const float* __restrict__ rowLoss,
                                                    const float* __restrict__ rowValid,
                                                    const float* __restrict__ perB,
                                                    float* __restrict__ out) {
    const int tid = threadIdx.x;
    __shared__ float sl[256], sv[256];
    float l = 0.0f, v = 0.0f;
    for (int i = tid; i < ROWS; i += 256) { l += rowLoss[i]; v += rowValid[i]; }
    sl[tid] = l; sv[tid] = v; __syncthreads();
    for (int o = 128; o; o >>= 1) {
        if (tid < o) { sl[tid] += sl[tid + o]; sv[tid] += sv[tid + o]; }
        __syncthreads();
    }
    if (tid == 0) {
        const float ce = sl[0] / fmaxf(sv[0], 1.0f);
        float rep = 0.0f;
        for (int b = 0; b < B_DIM; ++b) rep += perB[b];
        rep *= (1.0f / B_DIM);
        out[0] = ce + REP_W * rep;
        out[1] = ce;
        out[2] = rep;
    }
}

// ------------------------------- launcher -----------------------------------

extern "C" void kernel_launch(void* const* d_in, const int* in_sizes, int n_in,
                              void* d_out, int out_size, void* d_ws, size_t ws_size,
                              hipStream_t stream) {
    const float* logits = (const float*)d_in[0];
    const int*   labels = (const int*)d_in[1];
    // d_in[2] (input_ids) only gates rep != None in the reference; always present.
    float* out = (float*)d_out;

    float*    rowLoss  = (float*)d_ws;                       // [4096]
    float*    rowValid = rowLoss + ROWS;                     // [4096]
    unsigned* counts   = (unsigned*)(rowValid + ROWS);       // [8*32000]
    float*    perB     = (float*)(counts + (size_t)B_DIM * V_DIM);  // [8]

    zero_counts_kernel<<<(B_DIM * V_DIM + 255) / 256, 256, 0, stream>>>(counts);
    row_kernel<<<ROWS, 256, 0, stream>>>(logits, labels, rowLoss, rowValid, counts);
    rep_kernel<<<B_DIM, 256, 0, stream>>>(counts, perB);
    final_kernel<<<1, 256, 0, stream>>>(rowLoss, rowValid, perB, out);
}